// ConvKernel_34565896798380
// MI455X (gfx1250) — compile-verified
//
#include <hip/hip_runtime.h>
#include <math.h>

#define NQ 10
#define DIM 1024
#define BATCH 8192
#define PI_F 3.14159265358979323846f

typedef __attribute__((ext_vector_type(2))) float v2f;
typedef __attribute__((ext_vector_type(8))) float v8f;

__device__ __forceinline__ v8f wmma_f32x4(v2f a, v2f b, v8f c) {
  // 8 args: (neg_a, A, neg_b, B, c_mod, C, reuse_a, reuse_b)
  return __builtin_amdgcn_wmma_f32_16x16x4_f32(false, a, false, b, (short)0, c, false, false);
}

// Async memory->LDS copy (CDNA5): GVS mode, vdst = LDS byte offset,
// vaddr = per-lane 32-bit byte offset, saddr = 64-bit base. Tracked by ASYNCcnt.
__device__ __forceinline__ void async_b128(unsigned lds_off, unsigned goff, const float* base) {
  asm volatile("global_load_async_to_lds_b128 %0, %1, %2 offset:0"
               :: "v"(lds_off), "v"(goff), "s"(base) : "memory");
}
__device__ __forceinline__ void wait_async0() {
  asm volatile("s_wait_asynccnt 0x0" ::: "memory");
}

// K0: flag = (max(inputs) > 1)  -> whether to apply arctan
__global__ void k_flag(const float* __restrict__ in, int n, int* __restrict__ flag) {
  __shared__ float red[256];
  float m = -1e30f;
  for (int i = threadIdx.x; i < n; i += 256) m = fmaxf(m, in[i]);
  red[threadIdx.x] = m;
  __syncthreads();
  for (int s = 128; s > 0; s >>= 1) {
    if ((int)threadIdx.x < s) red[threadIdx.x] = fmaxf(red[threadIdx.x], red[threadIdx.x + s]);
    __syncthreads();
  }
  if (threadIdx.x == 0) flag[0] = (red[0] > 1.0f) ? 1 : 0;
}

// K1: extract perm f from E (E[f(j),j]==1), g=f^-1, per-qubit U=Rx@Rz, Rf=Ry^3
__global__ void k_pre(const float* __restrict__ w, const float* __restrict__ E,
                      float* __restrict__ Rf, float* __restrict__ U,
                      int* __restrict__ f, int* __restrict__ g) {
  int t = threadIdx.x;  // 1024 threads, one per column
  int row = 0;
  for (int i = 0; i < DIM; ++i)
    if (E[i * DIM + t] > 0.5f) row = i;
  f[t] = row;
  g[row] = t;  // bijection -> disjoint writes
  if (t < NQ) {
    float a  = PI_F * w[3 + t];        // Rx half-angle
    float bq = PI_F * w[3 + NQ + t];   // Rz half-angle
    float sa, ca, sb, cb;
    sincosf(a, &sa, &ca);
    sincosf(bq, &sb, &cb);
    float* u = &U[t * 8];
    u[0] =  ca * cb; u[1] = -ca * sb;  // U00 = ca*e^{-ib}
    u[2] =  sa * sb; u[3] = -sa * cb;  // U01 = -i*sa*e^{+ib}
    u[4] = -sa * sb; u[5] = -sa * cb;  // U10 = -i*sa*e^{-ib}
    u[6] =  ca * cb; u[7] =  ca * sb;  // U11 = ca*e^{+ib}
  }
  if (t == 0) {
    float phi = PI_F * (w[0] + w[1] + w[2]);  // Ry rotations add
    float sp, cp; sincosf(phi, &sp, &cp);
    Rf[0] = cp; Rf[1] = -sp; Rf[2] = sp; Rf[3] = cp;
  }
}

// K2: Bt[j*DIM + i] = M[i,j] = R[g(i), f(j)],  R = kron of U_q
__global__ void k_buildM(const float* __restrict__ U, const int* __restrict__ f,
                         const int* __restrict__ g, float* __restrict__ Btr,
                         float* __restrict__ Bti) {
  int tid = blockIdx.x * 256 + threadIdx.x;
  int i = tid & (DIM - 1);   // N index
  int j = tid >> 10;         // K index
  int a = g[i], b = f[j];
  float re = 1.0f, im = 0.0f;
#pragma unroll
  for (int q = 0; q < NQ; ++q) {
    int aq = (a >> (NQ - 1 - q)) & 1;
    int bq = (b >> (NQ - 1 - q)) & 1;
    const float* u = &U[q * 8 + (aq * 2 + bq) * 2];
    float ur = u[0], ui = u[1];
    float nr = re * ur - im * ui;
    float ni = re * ui + im * ur;
    re = nr; im = ni;
  }
  Btr[tid] = re;
  Bti[tid] = im;
}

// K3: psi[b,k] = (1/32) e^{i*pi*(-T + (T^2 - S2)/2)}
__global__ void k_psi(const float* __restrict__ in, const int* __restrict__ flag,
                      float* __restrict__ pr, float* __restrict__ pim) {
  int tid = blockIdx.x * 256 + threadIdx.x;
  int b = tid >> 10;
  int k = tid & (DIM - 1);
  bool doAtan = flag[0] != 0;
  float T = 0.0f, S2 = 0.0f;
#pragma unroll
  for (int q = 0; q < NQ; ++q) {
    float x = in[b * NQ + q];
    if (doAtan) x = atanf(x);
    float s = ((k >> (NQ - 1 - q)) & 1) ? -1.0f : 1.0f;
    T += x * s;
    S2 += x * x;
  }
  float phi = PI_F * (-T + 0.5f * (T * T - S2));
  float sn, cs;
  sincosf(phi, &sn, &cs);
  pr[tid]  = cs * 0.03125f;
  pim[tid] = sn * 0.03125f;
}

// K4: complex GEMM C[b,i] = sum_j psi[b,j]*Bt[j,i] for cols [n0,n0+64) and
// [n0+512,n0+576), fused Rf mix + |u1|^2-|u0|^2 row reduction -> partials.
// Staging uses global_load_async_to_lds_b128 (ASYNCcnt path).
__launch_bounds__(256)
__global__ void k_gemm(const float* __restrict__ pr, const float* __restrict__ pim,
                       const float* __restrict__ Btr, const float* __restrict__ Bti,
                       const float* __restrict__ Rf, float* __restrict__ partial) {
  // A rows padded to 20 dwords (80B: 16B-multiple for async B128, 8B for b64
  // fragment loads, 20*r mod 64 distinct for r=0..15 -> conflict-free).
  __shared__ __align__(16) float As0[64][20];
  __shared__ __align__(16) float As1[64][20];
  // B rows padded to 68 dwords (272B = 17*16: async-aligned; n contiguous).
  __shared__ __align__(16) float Bs[2][2][16][68];
  __shared__ float red[8][16];

  const int tid  = threadIdx.x;
  const int wave = tid >> 5;
  const int lane = tid & 31;
  const int wm = wave & 3;   // M subtile (16 rows each)
  const int wn = wave >> 2;  // N half (32 cols each)
  const int m0 = blockIdx.x * 64;
  const int n0 = blockIdx.y * 64;

  v8f acc[2][2][2] = {};  // [re/im][colblk low/high][nsub]

  const int arow = wm * 16 + (lane & 15);
  const int koff = (lane >> 4) * 2;  // K split: lanes 0-15 -> K0/1, 16-31 -> K2/3

  const int ar_r = tid >> 2, ar_c = (tid & 3) * 4;  // A staging: 64 rows x 4 xB128
  const int bk = tid >> 4, bc = (tid & 15) * 4;     // B staging: 16 rows x 16 xB128

  // Per-thread LDS destination byte offsets (constant across K loop).
  const unsigned ldsA0  = (unsigned)(uintptr_t)&As0[ar_r][ar_c];
  const unsigned ldsA1  = (unsigned)(uintptr_t)&As1[ar_r][ar_c];
  const unsigned ldsB00 = (unsigned)(uintptr_t)&Bs[0][0][bk][bc];
  const unsigned ldsB01 = (unsigned)(uintptr_t)&Bs[0][1][bk][bc];
  const unsigned ldsB10 = (unsigned)(uintptr_t)&Bs[1][0][bk][bc];
  const unsigned ldsB11 = (unsigned)(uintptr_t)&Bs[1][1][bk][bc];

  for (int k0 = 0; k0 < DIM; k0 += 16) {
    __syncthreads();  // previous iteration's LDS reads done
    {
      const unsigned ga = (unsigned)(((m0 + ar_r) * DIM + k0 + ar_c) * 4);
      async_b128(ldsA0, ga, pr);
      async_b128(ldsA1, ga, pim);
      const unsigned gbL = (unsigned)((((k0 + bk) * DIM) + n0 + bc) * 4);
      const unsigned gbH = gbL + 512u * 4u;  // paired column block (+512 cols)
      async_b128(ldsB00, gbL, Btr);
      async_b128(ldsB01, gbH, Btr);
      async_b128(ldsB10, gbL, Bti);
      async_b128(ldsB11, gbH, Bti);
      wait_async0();  // all this wave's async LDS writes landed
    }
    __syncthreads();  // all waves' tiles visible
#pragma unroll
    for (int kk = 0; kk < 16; kk += 4) {
      v2f ar = *(const v2f*)&As0[arow][kk + koff];
      v2f ai = *(const v2f*)&As1[arow][kk + koff];
      v2f nai; nai.x = -ai.x; nai.y = -ai.y;  // f32 WMMA has no A-neg modifier
#pragma unroll
      for (int ns = 0; ns < 2; ++ns) {
        const int bcol = wn * 32 + ns * 16 + (lane & 15);
#pragma unroll
        for (int cb = 0; cb < 2; ++cb) {
          v2f br, bi;
          br.x = Bs[0][cb][kk + koff][bcol]; br.y = Bs[0][cb][kk + koff + 1][bcol];
          bi.x = Bs[1][cb][kk + koff][bcol]; bi.y = Bs[1][cb][kk + koff + 1][bcol];
          acc[0][cb][ns] = wmma_f32x4(ar,  br, acc[0][cb][ns]);  // Cr += Ar*Br
          acc[0][cb][ns] = wmma_f32x4(nai, bi, acc[0][cb][ns]);  // Cr -= Ai*Bi
          acc[1][cb][ns] = wmma_f32x4(ar,  bi, acc[1][cb][ns]);  // Ci += Ar*Bi
          acc[1][cb][ns] = wmma_f32x4(ai,  br, acc[1][cb][ns]);  // Ci += Ai*Br
        }
      }
    }
  }

  // Epilogue: u = Rf (x) I_512 applied to (low,high) column pair; reduce cols.
  const float r00 = Rf[0], r01 = Rf[1], r10 = Rf[2], r11 = Rf[3];
#pragma unroll
  for (int v = 0; v < 8; ++v) {  // C layout: VGPR v -> row v (lanes 0-15), v+8 (16-31)
    float val = 0.0f;
#pragma unroll
    for (int ns = 0; ns < 2; ++ns) {
      float crl = acc[0][0][ns][v], cil = acc[1][0][ns][v];
      float crh = acc[0][1][ns][v], cih = acc[1][1][ns][v];
      float u0r = r00 * crl + r01 * crh, u0i = r00 * cil + r01 * cih;
      float u1r = r10 * crl + r11 * crh, u1i = r10 * cil + r11 * cih;
      val += u1r * u1r + u1i * u1i - u0r * u0r - u0i * u0i;
    }
    for (int off = 8; off > 0; off >>= 1) val += __shfl_xor(val, off, 32);
    if ((lane & 15) == 0) red[wave][v + (lane >> 4) * 8] = val;
  }
  __syncthreads();
  if (tid < 64) {
    int r = tid;
    float s = red[r >> 4][r & 15] + red[(r >> 4) + 4][r & 15];
    partial[(size_t)(m0 + r) * 8 + blockIdx.y] = s;
  }
}

// K5: out[b] = sum of 8 column-block partials (deterministic)
__global__ void k_final(const float* __restrict__ partial, float* __restrict__ out) {
  int b = blockIdx.x * 256 + threadIdx.x;
  float s = 0.0f;
#pragma unroll
  for (int c = 0; c < 8; ++c) s += partial[(size_t)b * 8 + c];
  out[b] = s;
}

extern "C" void kernel_launch(void* const* d_in, const int* in_sizes, int n_in,
                              void* d_out, int out_size, void* d_ws, size_t ws_size,
                              hipStream_t stream) {
  const float* inputs = (const float*)d_in[0];  // (8192,10) f32
  const float* weight = (const float*)d_in[1];  // (23,) f32
  const float* Emat   = (const float*)d_in[2];  // (1024,1024) f32
  float* out = (float*)d_out;                   // (8192,) f32

  char* ws = (char*)d_ws;
  int*   flag  = (int*)ws;
  float* Rf    = (float*)(ws + 256);
  float* U     = (float*)(ws + 512);
  int*   fperm = (int*)(ws + 4096);
  int*   gperm = (int*)(ws + 8192);
  float* psir  = (float*)(ws + 16384);
  float* psii  = psir + (size_t)BATCH * DIM;
  float* Btr   = psii + (size_t)BATCH * DIM;
  float* Bti   = Btr + (size_t)DIM * DIM;
  float* partial = Bti + (size_t)DIM * DIM;  // BATCH*8 floats

  k_flag<<<1, 256, 0, stream>>>(inputs, BATCH * NQ, flag);
  k_pre<<<1, 1024, 0, stream>>>(weight, Emat, Rf, U, fperm, gperm);
  k_buildM<<<(DIM * DIM) / 256, 256, 0, stream>>>(U, fperm, gperm, Btr, Bti);
  k_psi<<<(BATCH * DIM) / 256, 256, 0, stream>>>(inputs, flag, psir, psii);
  dim3 g4(BATCH / 64, 8);
  k_gemm<<<g4, 256, 0, stream>>>(psir, psii, Btr, Bti, Rf, partial);
  k_final<<<BATCH / 256, 256, 0, stream>>>(partial, out);
}